// DiKT_76218489635287
// MI455X (gfx1250) — compile-verified
//
#include <hip/hip_runtime.h>
#include <math.h>

typedef __attribute__((ext_vector_type(2))) float v2f;
typedef __attribute__((ext_vector_type(8))) float v8f;

#define BATCH 128
#define SEQ   128
#define VDIM  128
#define KDIM  128
#define CDIM  64
#define QNUM  10000
#define NROWS (BATCH * SEQ)

__device__ __forceinline__ float sigmoidf_(float x) {
    return 1.0f / (1.0f + __expf(-x));
}

// ---------------------------------------------------------------------------
// Merged erase/add projections for right+wrong streams.
// grid = (NROWS/16, 4); grid.y: bit0 = op (0 erase/sigmoid, 1 add/tanh),
//                               bit1 = stream (0 right, 1 wrong).
// Block = 256 (8 waves); wave w owns output columns [16w, 16w+16).
// FP32 WMMA: D(16x16) += A(16x4) * B(4x16), K-loop of 32 iterations.
//   A lane map: lane L -> m = L%16, a[j] = X[m][2*(L/16)+j]
//   B lane map: lane L -> n = L%16, b[j] = W[n][2*(L/16)+j]  (B = W^T tile)
//   D lane map: acc[r] -> element (m = r + 8*(L/16), n = L%16)
// ---------------------------------------------------------------------------
__global__ __launch_bounds__(256) void gemm_all(
    const int* __restrict__ right_in, const int* __restrict__ wrong_in,
    const float* __restrict__ emb,
    const float* __restrict__ erase_W, const float* __restrict__ erase_b,
    const float* __restrict__ add_W,   const float* __restrict__ add_b,
    float* __restrict__ E_r, float* __restrict__ A_r,
    float* __restrict__ E_w, float* __restrict__ A_w)
{
    const int job = blockIdx.y;
    const int act = job & 1;
    const int* __restrict__ ids   = (job & 2) ? wrong_in : right_in;
    const float* __restrict__ W    = act ? add_W : erase_W;
    const float* __restrict__ bias = act ? add_b : erase_b;
    float* __restrict__ out = (job == 0) ? E_r : (job == 1) ? A_r
                            : (job == 2) ? E_w : A_w;

    const int tid  = threadIdx.x;
    const int wave = tid >> 5;
    const int lane = tid & 31;
    const int mr   = lane & 15;   // m for A, n for B
    const int hi   = lane >> 4;   // 0/1: K sub-pair select
    const int row0 = blockIdx.x * 16;
    const int col0 = wave * 16;

    const long id = (long)ids[row0 + mr];
    const float* __restrict__ xrow = emb + id * KDIM;
    const float* __restrict__ wrow = W + (long)(col0 + mr) * KDIM;

    v8f acc = {};
    #pragma unroll 4
    for (int k = 0; k < KDIM; k += 4) {
        const int koff = k + 2 * hi;
        v2f a = *(const v2f*)(xrow + koff);
        v2f b = *(const v2f*)(wrow + koff);
        acc = __builtin_amdgcn_wmma_f32_16x16x4_f32(
            false, a, false, b, (short)0, acc, false, false);
    }

    const int n = col0 + mr;
    const float bn = bias[n];
    #pragma unroll
    for (int r = 0; r < 8; ++r) {
        const int m = r + 8 * hi;
        float v = acc[r] + bn;
        v = (act == 0) ? sigmoidf_(v) : tanhf(v);
        out[(long)(row0 + m) * VDIM + n] = v;
    }
}

// ---------------------------------------------------------------------------
// Merged correlation softmax: out[r][c] = softmax_c( q_emb[qid(r)] . key_W[c] )
// qid = id - Q*(id > Q) (identity for target ids <= Q).
// grid.x = 2*(NROWS/16) + BATCH/16 blocks, block = 128 (4 waves, 64 cols).
// ---------------------------------------------------------------------------
__global__ __launch_bounds__(128) void corr_all(
    const int* __restrict__ right_in, const int* __restrict__ wrong_in,
    const int* __restrict__ target_id,
    const float* __restrict__ q_emb, const float* __restrict__ key_W,
    float* __restrict__ W_r, float* __restrict__ W_w, float* __restrict__ w_t)
{
    __shared__ float tile[16][CDIM + 4];

    const int bx = blockIdx.x;
    const int* __restrict__ ids;
    float* __restrict__ out;
    int row0;
    if (bx < NROWS / 16)            { ids = right_in;  out = W_r; row0 = bx * 16; }
    else if (bx < 2 * (NROWS / 16)) { ids = wrong_in;  out = W_w; row0 = (bx - NROWS / 16) * 16; }
    else                            { ids = target_id; out = w_t; row0 = (bx - 2 * (NROWS / 16)) * 16; }

    const int tid  = threadIdx.x;
    const int wave = tid >> 5;
    const int lane = tid & 31;
    const int mr   = lane & 15;
    const int hi   = lane >> 4;
    const int col0 = wave * 16;

    int id = ids[row0 + mr];
    if (id > QNUM) id -= QNUM;
    const float* __restrict__ xrow = q_emb + (long)id * KDIM;
    const float* __restrict__ wrow = key_W + (long)(col0 + mr) * KDIM;

    v8f acc = {};
    #pragma unroll 4
    for (int k = 0; k < KDIM; k += 4) {
        const int koff = k + 2 * hi;
        v2f a = *(const v2f*)(xrow + koff);
        v2f b = *(const v2f*)(wrow + koff);
        acc = __builtin_amdgcn_wmma_f32_16x16x4_f32(
            false, a, false, b, (short)0, acc, false, false);
    }
    #pragma unroll
    for (int r = 0; r < 8; ++r)
        tile[r + 8 * hi][col0 + mr] = acc[r];
    __syncthreads();

    if (tid < 16) {
        float mx = -3.0e38f;
        #pragma unroll 4
        for (int c = 0; c < CDIM; ++c) mx = fmaxf(mx, tile[tid][c]);
        float s = 0.0f;
        #pragma unroll 4
        for (int c = 0; c < CDIM; ++c) {
            float e = __expf(tile[tid][c] - mx);
            tile[tid][c] = e;
            s += e;
        }
        const float inv = 1.0f / s;
        float* __restrict__ orow = out + (long)(row0 + tid) * CDIM;
        #pragma unroll 4
        for (int c = 0; c < CDIM; ++c) orow[c] = tile[tid][c] * inv;
    }
}

// ---------------------------------------------------------------------------
// Merged sequential memory scans (right + wrong in one launch):
//   m[v][c] <- m + w[c]*(a[v] - e[v]*m), 128 steps.
// grid.x = 2*BATCH; blockIdx.x>>7 selects the memory stream.
// Block = 256. Thread owns (v = tid&127, c in [32*(tid>>7), +32)).
// Double-buffered LDS + register prefetch of step s+1 issued before the
// single per-step barrier, so global latency overlaps barrier + compute.
// ---------------------------------------------------------------------------
__global__ __launch_bounds__(256) void scan_all(
    const float* __restrict__ E_r, const float* __restrict__ A_r,
    const float* __restrict__ W_r, const float* __restrict__ m0_r,
    float* __restrict__ mo_r,
    const float* __restrict__ E_w, const float* __restrict__ A_w,
    const float* __restrict__ W_w, const float* __restrict__ m0_w,
    float* __restrict__ mo_w)
{
    __shared__ float se[2][VDIM], sa[2][VDIM], sw[2][CDIM];

    const int sel = blockIdx.x >> 7;
    const int b   = blockIdx.x & (BATCH - 1);
    const float* __restrict__ E  = sel ? E_w  : E_r;
    const float* __restrict__ A  = sel ? A_w  : A_r;
    const float* __restrict__ Wc = sel ? W_w  : W_r;
    const float* __restrict__ mi = sel ? m0_w : m0_r;
    float* __restrict__ mo       = sel ? mo_w : mo_r;

    const int tid = threadIdx.x;
    const int v   = tid & 127;
    const int c0  = (tid >> 7) * 32;

    float m[32];
    #pragma unroll
    for (int i = 0; i < 32; ++i) m[i] = mi[v * CDIM + c0 + i];

    const float* __restrict__ Eb = E  + (long)b * SEQ * VDIM;
    const float* __restrict__ Ab = A  + (long)b * SEQ * VDIM;
    const float* __restrict__ Wb = Wc + (long)b * SEQ * CDIM;

    // preload step 0 into buffer 0
    if (tid < VDIM) {
        se[0][tid] = Eb[tid];
        sa[0][tid] = Ab[tid];
    } else if (tid < VDIM + CDIM) {
        sw[0][tid - VDIM] = Wb[tid - VDIM];
    }

    int p = 0;
    for (int s = 0; s < SEQ; ++s) {
        // issue next step's loads before the barrier (latency overlap)
        float ne = 0.0f, na = 0.0f, nw = 0.0f;
        if (s + 1 < SEQ) {
            if (tid < VDIM) {
                ne = Eb[(s + 1) * VDIM + tid];
                na = Ab[(s + 1) * VDIM + tid];
            } else if (tid < VDIM + CDIM) {
                nw = Wb[(s + 1) * CDIM + (tid - VDIM)];
            }
        }
        __syncthreads();   // buf[p] (written last iteration) is now visible
        const float e = se[p][v];
        const float a = sa[p][v];
        #pragma unroll
        for (int i = 0; i < 32; ++i) {
            const float t = fmaf(-e, m[i], a);   // a - e*m
            m[i] = fmaf(sw[p][c0 + i], t, m[i]); // m + w*(a - e*m)
        }
        // stage step s+1 into the other buffer
        if (tid < VDIM) {
            se[p ^ 1][tid] = ne;
            sa[p ^ 1][tid] = na;
        } else if (tid < VDIM + CDIM) {
            sw[p ^ 1][tid - VDIM] = nw;
        }
        p ^= 1;
    }

    #pragma unroll
    for (int i = 0; i < 32; ++i) mo[v * CDIM + c0 + i] = m[i];
}

// ---------------------------------------------------------------------------
// sig[0] = sigmoid(#nonzero right), sig[1] = sigmoid(#nonzero wrong)
// ---------------------------------------------------------------------------
__global__ __launch_bounds__(256) void count_kernel(
    const int* __restrict__ ri, const int* __restrict__ wi,
    float* __restrict__ sig)
{
    __shared__ float sr[256], sf[256];
    const int tid = threadIdx.x;
    float cr = 0.0f, cf = 0.0f;
    for (int i = tid; i < NROWS; i += 256) {
        cr += (ri[i] > 0) ? 1.0f : 0.0f;   // clip(x,0,1), x >= 0
        cf += (wi[i] > 0) ? 1.0f : 0.0f;
    }
    sr[tid] = cr; sf[tid] = cf;
    __syncthreads();
    for (int st = 128; st > 0; st >>= 1) {
        if (tid < st) { sr[tid] += sr[tid + st]; sf[tid] += sf[tid + st]; }
        __syncthreads();
    }
    if (tid == 0) {
        sig[0] = sigmoidf_(sr[0]);
        sig[1] = sigmoidf_(sf[0]);
    }
}

// ---------------------------------------------------------------------------
// Readout: grid.x = BATCH, block = 128.
// ---------------------------------------------------------------------------
__global__ __launch_bounds__(128) void readout_kernel(
    const float* __restrict__ mem_r, const float* __restrict__ mem_w,
    const float* __restrict__ w_t, const int* __restrict__ target_id,
    const float* __restrict__ q_emb,
    const float* __restrict__ rsum_W, const float* __restrict__ rsum_b,
    const float* __restrict__ wsum_W, const float* __restrict__ wsum_b,
    const float* __restrict__ succ_W, const float* __restrict__ succ_b,
    const float* __restrict__ fail_W, const float* __restrict__ fail_b,
    const float* __restrict__ diff_W, const float* __restrict__ diff_b,
    const float* __restrict__ sig, float* __restrict__ out)
{
    __shared__ float wt[CDIM], qv[KDIM], rrd[VDIM], wrd[VDIM];
    __shared__ float rs[128], ws_[128];
    __shared__ float red0[128], red1[128], red2[128];

    const int b = blockIdx.x;
    const int tid = threadIdx.x;

    if (tid < CDIM) wt[tid] = w_t[b * CDIM + tid];
    qv[tid] = q_emb[(long)target_id[b] * KDIM + tid];
    __syncthreads();

    // memory reads: rr[v] = sum_c mem[b][v][c] * wt[c]
    {
        const float* __restrict__ mr = mem_r + (long)b * VDIM * CDIM + tid * CDIM;
        const float* __restrict__ mw = mem_w + (long)b * VDIM * CDIM + tid * CDIM;
        float rr = 0.0f, wr = 0.0f;
        #pragma unroll 4
        for (int c = 0; c < CDIM; ++c) {
            rr = fmaf(mr[c], wt[c], rr);
            wr = fmaf(mw[c], wt[c], wr);
        }
        rrd[tid] = rr; wrd[tid] = wr;
    }
    __syncthreads();

    // summary layers: tanh([read, qv] @ Wsum^T + b)
    {
        float sr = rsum_b[tid], sw = wsum_b[tid];
        const float* __restrict__ rW = rsum_W + tid * 256;
        const float* __restrict__ wW = wsum_W + tid * 256;
        #pragma unroll 4
        for (int k = 0; k < 128; ++k) {
            sr = fmaf(rrd[k], rW[k], sr);
            sr = fmaf(qv[k], rW[128 + k], sr);
            sw = fmaf(wrd[k], wW[k], sw);
            sw = fmaf(qv[k], wW[128 + k], sw);
        }
        rs[tid] = tanhf(sr);
        ws_[tid] = tanhf(sw);
    }
    __syncthreads();

    red0[tid] = rs[tid] * succ_W[tid];
    red1[tid] = ws_[tid] * fail_W[tid];
    red2[tid] = qv[tid] * diff_W[tid];
    __syncthreads();
    for (int st = 64; st > 0; st >>= 1) {
        if (tid < st) {
            red0[tid] += red0[tid + st];
            red1[tid] += red1[tid + st];
            red2[tid] += red2[tid + st];
        }
        __syncthreads();
    }
    if (tid == 0) {
        const float sl = tanhf(red0[0] + succ_b[0]);
        const float fl = tanhf(red1[0] + fail_b[0]);
        const float dl = tanhf(red2[0] + diff_b[0]);
        out[b] = sl * sig[0] + fl * sig[1] - 2.0f * dl;
    }
}

// ---------------------------------------------------------------------------
extern "C" void kernel_launch(void* const* d_in, const int* in_sizes, int n_in,
                              void* d_out, int out_size, void* d_ws, size_t ws_size,
                              hipStream_t stream)
{
    const int*   right_in  = (const int*)  d_in[0];
    const int*   wrong_in  = (const int*)  d_in[1];
    const int*   target_id = (const int*)  d_in[2];
    const float* q_emb     = (const float*)d_in[3];
    const float* i_emb     = (const float*)d_in[4];
    const float* key_W     = (const float*)d_in[5];
    const float* erase_W   = (const float*)d_in[6];
    const float* erase_b   = (const float*)d_in[7];
    const float* add_W     = (const float*)d_in[8];
    const float* add_b     = (const float*)d_in[9];
    const float* rsum_W    = (const float*)d_in[10];
    const float* rsum_b    = (const float*)d_in[11];
    const float* wsum_W    = (const float*)d_in[12];
    const float* wsum_b    = (const float*)d_in[13];
    const float* succ_W    = (const float*)d_in[14];
    const float* succ_b    = (const float*)d_in[15];
    const float* fail_W    = (const float*)d_in[16];
    const float* fail_b    = (const float*)d_in[17];
    const float* diff_W    = (const float*)d_in[18];
    const float* diff_b    = (const float*)d_in[19];
    const float* r_mem0    = (const float*)d_in[20];
    const float* w_mem0    = (const float*)d_in[21];

    float* ws = (float*)d_ws;
    const size_t N_EA  = (size_t)NROWS * VDIM;   // 2,097,152
    const size_t N_W   = (size_t)NROWS * CDIM;   // 1,048,576
    const size_t N_MEM = (size_t)BATCH * VDIM * CDIM;

    float* E_r   = ws;               ws += N_EA;
    float* A_r   = ws;               ws += N_EA;
    float* E_w   = ws;               ws += N_EA;
    float* A_w   = ws;               ws += N_EA;
    float* W_r   = ws;               ws += N_W;
    float* W_w   = ws;               ws += N_W;
    float* mem_r = ws;               ws += N_MEM;
    float* mem_w = ws;               ws += N_MEM;
    float* w_t   = ws;               ws += (size_t)BATCH * CDIM;
    float* sig   = ws;               ws += 2;

    // scalar sigmoid(counts)
    count_kernel<<<1, 256, 0, stream>>>(right_in, wrong_in, sig);

    // all four erase/add projections in one launch (FP32 WMMA GEMMs)
    gemm_all<<<dim3(NROWS / 16, 4), 256, 0, stream>>>(
        right_in, wrong_in, i_emb, erase_W, erase_b, add_W, add_b,
        E_r, A_r, E_w, A_w);

    // all three correlation-softmax jobs in one launch
    corr_all<<<dim3(2 * (NROWS / 16) + BATCH / 16), 128, 0, stream>>>(
        right_in, wrong_in, target_id, q_emb, key_W, W_r, W_w, w_t);

    // both sequential memory scans in one launch
    scan_all<<<dim3(2 * BATCH), 256, 0, stream>>>(
        E_r, A_r, W_r, r_mem0, mem_r,
        E_w, A_w, W_w, w_mem0, mem_w);

    // readout heads
    readout_kernel<<<dim3(BATCH), 128, 0, stream>>>(
        mem_r, mem_w, w_t, target_id, q_emb,
        rsum_W, rsum_b, wsum_W, wsum_b,
        succ_W, succ_b, fail_W, fail_b, diff_W, diff_b,
        sig, (float*)d_out);
}